// GraphSAGE_27693949125044
// MI455X (gfx1250) — compile-verified
//
#include <hip/hip_runtime.h>
#include <hip/hip_bf16.h>

typedef __attribute__((ext_vector_type(2))) float v2f;
typedef __attribute__((ext_vector_type(8))) float v8f;

#define N_NODES 100000
#define N_EDGES 1600000
#define FDIM 64
#define N_GRAPHS 64

__global__ void zero_kernel(float* __restrict__ p, int n) {
    int i = blockIdx.x * blockDim.x + threadIdx.x;
    if (i < n) p[i] = 0.0f;
}

__global__ void deg_kernel(const int* __restrict__ dst, float* __restrict__ cnt) {
    int e = blockIdx.x * blockDim.x + threadIdx.x;
    if (e < N_EDGES) atomicAdd(&cnt[dst[e]], 1.0f);
}

__global__ void inv_kernel(const float* __restrict__ cnt, float* __restrict__ inv) {
    int i = blockIdx.x * blockDim.x + threadIdx.x;
    if (i < N_NODES) inv[i] = 1.0f / fmaxf(cnt[i], 1.0f);
}

// one thread per (edge, feature); 64 consecutive lanes share one edge -> coalesced row loads
__global__ void scatter_kernel(const float* __restrict__ h,
                               const int* __restrict__ src,
                               const int* __restrict__ dst,
                               float* __restrict__ agg) {
    int gid = blockIdx.x * blockDim.x + threadIdx.x;   // exactly E*64 threads
    int e = gid >> 6;
    int f = gid & 63;
    int s = src[e];
    int d = dst[e];
    atomicAdd(&agg[d * FDIM + f], h[s * FDIM + f]);
}

// hout[16x16 tile] = act( (agg*inv) @ Wl + bias + hin @ Wr ) via V_WMMA_F32_16X16X4_F32
// one wave per tile; 8 waves (256 threads) per block; grid covers 6250x4 tiles exactly
__global__ void sage_gemm_kernel(const float* __restrict__ agg,
                                 const float* __restrict__ hin,
                                 const float* __restrict__ inv,
                                 const float* __restrict__ Wl,
                                 const float* __restrict__ bias,
                                 const float* __restrict__ Wr,
                                 float* __restrict__ hout,
                                 int do_relu) {
    int wave = (blockIdx.x * blockDim.x + threadIdx.x) >> 5;  // global wave id
    int lane = threadIdx.x & 31;
    int rowBase = (wave >> 2) << 4;     // tile row * 16
    int colBase = (wave & 3) << 4;      // tile col * 16
    int r  = lane & 15;                 // A-row within tile / B-column / C-column
    int hi = lane >> 4;                 // half-wave: K pair select (0->K0,K1 ; 1->K2,K3)

    float scale = inv[rowBase + r];
    const float* arow = agg + (rowBase + r) * FDIM;
    const float* hrow = hin + (rowBase + r) * FDIM;

    v8f c = {};

    // agg(mean) @ Wl : K = 64 in 16 steps of 4
    #pragma unroll
    for (int k0 = 0; k0 < FDIM; k0 += 4) {
        int ka = k0 + 2 * hi;
        v2f a, b;
        a.x = arow[ka]     * scale;
        a.y = arow[ka + 1] * scale;
        b.x = Wl[ka * FDIM + colBase + r];
        b.y = Wl[(ka + 1) * FDIM + colBase + r];
        c = __builtin_amdgcn_wmma_f32_16x16x4_f32(false, a, false, b,
                                                  (short)0, c, false, false);
    }
    // hin @ Wr
    #pragma unroll
    for (int k0 = 0; k0 < FDIM; k0 += 4) {
        int ka = k0 + 2 * hi;
        v2f a, b;
        a.x = hrow[ka];
        a.y = hrow[ka + 1];
        b.x = Wr[ka * FDIM + colBase + r];
        b.y = Wr[(ka + 1) * FDIM + colBase + r];
        c = __builtin_amdgcn_wmma_f32_16x16x4_f32(false, a, false, b,
                                                  (short)0, c, false, false);
    }

    float bv = bias[colBase + r];
    #pragma unroll
    for (int v = 0; v < 8; ++v) {
        float val = c[v] + bv;
        if (do_relu) val = fmaxf(val, 0.0f);
        int row = rowBase + (hi ? (v + 8) : v);   // C layout: M = vgpr (+8 upper half)
        hout[row * FDIM + colBase + r] = val;
    }
}

__global__ void pool_feat_kernel(const float* __restrict__ h,
                                 const int* __restrict__ batch,
                                 float* __restrict__ gs) {
    int gid = blockIdx.x * blockDim.x + threadIdx.x;   // exactly N*64 threads
    int node = gid >> 6;
    int f = gid & 63;
    atomicAdd(&gs[batch[node] * FDIM + f], h[gid]);
}

__global__ void pool_cnt_kernel(const int* __restrict__ batch, float* __restrict__ gcnt) {
    int i = blockIdx.x * blockDim.x + threadIdx.x;
    if (i < N_NODES) atomicAdd(&gcnt[batch[i]], 1.0f);
}

// t[g][j] = (gs[g]/cnt[g]) @ Wlin + blin   : 64 blocks x 256 threads
__global__ void head1_kernel(const float* __restrict__ gs,
                             const float* __restrict__ gcnt,
                             const float* __restrict__ Wlin,
                             const float* __restrict__ blin,
                             float* __restrict__ t) {
    int gi = blockIdx.x;       // 0..63
    int j  = threadIdx.x;      // 0..255
    float invg = 1.0f / fmaxf(gcnt[gi], 1.0f);
    float acc = blin[j];
    #pragma unroll 4
    for (int k = 0; k < FDIM; ++k)
        acc += gs[gi * FDIM + k] * invg * Wlin[k * 256 + j];
    t[gi * 256 + j] = acc;
}

// batchnorm over the 64 graphs (training-mode batch stats, weight=1 bias=0) + relu, in place
__global__ void bn_relu_kernel(float* __restrict__ t) {
    int j = threadIdx.x;       // 256 columns
    float s = 0.0f, ss = 0.0f;
    for (int g = 0; g < N_GRAPHS; ++g) {
        float v = t[g * 256 + j];
        s  += v;
        ss += v * v;
    }
    float mu   = s * (1.0f / N_GRAPHS);
    float var  = ss * (1.0f / N_GRAPHS) - mu * mu;
    float rstd = rsqrtf(var + 1e-5f);
    for (int g = 0; g < N_GRAPHS; ++g) {
        float v = (t[g * 256 + j] - mu) * rstd;
        t[g * 256 + j] = fmaxf(v, 0.0f);
    }
}

__global__ void head2_kernel(const float* __restrict__ t,
                             const float* __restrict__ W2,
                             const float* __restrict__ b2,
                             float* __restrict__ out) {
    int gid = threadIdx.x;     // 128 threads: (graph, class)
    int g = gid >> 1;
    int cc = gid & 1;
    float acc = b2[cc];
    for (int j = 0; j < 256; ++j)
        acc += t[g * 256 + j] * W2[j * 2 + cc];
    out[g * 2 + cc] = acc;
}

extern "C" void kernel_launch(void* const* d_in, const int* in_sizes, int n_in,
                              void* d_out, int out_size, void* d_ws, size_t ws_size,
                              hipStream_t stream) {
    (void)in_sizes; (void)n_in; (void)out_size; (void)ws_size;

    const float* x     = (const float*)d_in[0];
    const int*   ei    = (const int*)d_in[1];
    const int*   src   = ei;
    const int*   dst   = ei + N_EDGES;
    // d_in[2] = edge_weight: unused by reference
    const int*   batch = (const int*)d_in[3];
    const float* W1l = (const float*)d_in[4];
    const float* b1  = (const float*)d_in[5];
    const float* W1r = (const float*)d_in[6];
    const float* W2l = (const float*)d_in[7];
    const float* b2  = (const float*)d_in[8];
    const float* W2r = (const float*)d_in[9];
    const float* W3l = (const float*)d_in[10];
    const float* b3  = (const float*)d_in[11];
    const float* W3r = (const float*)d_in[12];
    const float* Wlin  = (const float*)d_in[13];
    const float* blin  = (const float*)d_in[14];
    const float* Wlin2 = (const float*)d_in[15];
    const float* blin2 = (const float*)d_in[16];
    float* out = (float*)d_out;

    // workspace layout (floats)
    float* ws = (float*)d_ws;
    const int NF = N_NODES * FDIM;          // 6,400,000
    float* h1   = ws;
    float* h2   = h1 + NF;
    float* agg  = h2 + NF;
    float* cnt  = agg + NF;
    float* inv  = cnt + N_NODES;
    float* gs   = inv + N_NODES;            // [64 x 64]
    float* gcnt = gs + N_GRAPHS * FDIM;     // [64]
    float* t    = gcnt + N_GRAPHS;          // [64 x 256]

    const int B = 256;
    const int nodeBlocks    = (N_NODES + B - 1) / B;
    const int edgeBlocks    = N_EDGES / B;                  // 6250 exact
    const int scatterBlocks = (N_EDGES / B) * FDIM;         // 400000 exact
    const int zeroNFBlocks  = NF / B;                       // 25000 exact
    const int gemmBlocks    = (N_NODES / 16) * 4 / 8;       // 3125: 25000 tiles / 8 waves

    // degrees (shared by all three convs)
    zero_kernel<<<nodeBlocks, B, 0, stream>>>(cnt, N_NODES);
    deg_kernel<<<edgeBlocks, B, 0, stream>>>(dst, cnt);
    inv_kernel<<<nodeBlocks, B, 0, stream>>>(cnt, inv);

    // conv1: x -> h1 (relu)
    zero_kernel<<<zeroNFBlocks, B, 0, stream>>>(agg, NF);
    scatter_kernel<<<scatterBlocks, B, 0, stream>>>(x, src, dst, agg);
    sage_gemm_kernel<<<gemmBlocks, B, 0, stream>>>(agg, x, inv, W1l, b1, W1r, h1, 1);

    // conv2: h1 -> h2 (relu)
    zero_kernel<<<zeroNFBlocks, B, 0, stream>>>(agg, NF);
    scatter_kernel<<<scatterBlocks, B, 0, stream>>>(h1, src, dst, agg);
    sage_gemm_kernel<<<gemmBlocks, B, 0, stream>>>(agg, h1, inv, W2l, b2, W2r, h2, 1);

    // conv3: h2 -> h1 (no relu)
    zero_kernel<<<zeroNFBlocks, B, 0, stream>>>(agg, NF);
    scatter_kernel<<<scatterBlocks, B, 0, stream>>>(h2, src, dst, agg);
    sage_gemm_kernel<<<gemmBlocks, B, 0, stream>>>(agg, h2, inv, W3l, b3, W3r, h1, 0);

    // global mean pool
    zero_kernel<<<(N_GRAPHS * FDIM + N_GRAPHS + B - 1) / B, B, 0, stream>>>(
        gs, N_GRAPHS * FDIM + N_GRAPHS);    // zeros gs and gcnt (contiguous)
    pool_feat_kernel<<<zeroNFBlocks, B, 0, stream>>>(h1, batch, gs);
    pool_cnt_kernel<<<nodeBlocks, B, 0, stream>>>(batch, gcnt);

    // MLP head + batchnorm + classifier
    head1_kernel<<<N_GRAPHS, 256, 0, stream>>>(gs, gcnt, Wlin, blin, t);
    bn_relu_kernel<<<1, 256, 0, stream>>>(t);
    head2_kernel<<<1, 128, 0, stream>>>(t, Wlin2, blin2, out);
}